// Mamba_3779571220939
// MI455X (gfx1250) — compile-verified
//
#include <hip/hip_runtime.h>

typedef float v2f __attribute__((ext_vector_type(2)));
typedef float v8f __attribute__((ext_vector_type(8)));
typedef unsigned int u32;
typedef u32 u32x4 __attribute__((ext_vector_type(4)));
typedef u32 u32x8 __attribute__((ext_vector_type(8)));

#define LSEQ 128   // sequence length
#define NE   384   // n_embd
#define DI   768   // d_inner
#define NS   768   // d_state
#define RR   24    // dt_rank
#define NLAY 2
#define VOC  32000
#define KC   4     // conv kernel width
#define KB   32    // GEMM K-chunk staged in LDS

// ---------------------------------------------------------------------------
// Tensor Data Mover: 1-row load of `nelem` fp32 from global to LDS.
// D# packed per CDNA5 ISA:
//  group0: count=1, lds_addr, global_addr[56:0], type=2
//  group1: workgroup_mask=0 (not in cluster), data_size=2 (4B),
//          tensor_dim0=nelem, tensor_dim1=1, tile_dim0=nelem, tile_dim1=1,
//          tensor_dim0_stride=nelem, no pad/iterate/gather.
// Tracked by TENSORcnt; caller waits with s_wait_tensorcnt.
// ---------------------------------------------------------------------------
__device__ __forceinline__ void tdm_load_row_f32(const float* gsrc, u32 lds_off,
                                                 int nelem) {
  unsigned long long ga = (unsigned long long)gsrc;
  u32x4 g0;
  g0[0] = 1u;                                            // count=1
  g0[1] = lds_off;                                       // lds_addr
  g0[2] = (u32)ga;                                       // global_addr[31:0]
  g0[3] = ((u32)(ga >> 32) & 0x01ffffffu) | 0x80000000u; // ga[56:32] | type=2
  u32x8 g1;
  g1[0] = 0x00020000u;                                   // data_size=2 (4B)
  g1[1] = ((u32)nelem & 0xffffu) << 16;                  // tensor_dim0[15:0]
  g1[2] = (((u32)nelem >> 16) & 0xffffu) | (1u << 16);   // dim0[31:16]|dim1=1
  g1[3] = ((u32)nelem & 0xffffu) << 16;                  // tile_dim0
  g1[4] = 1u;                                            // tile_dim1=1, tile_dim2=0
  g1[5] = (u32)nelem;                                    // dim0_stride[31:0]
  g1[6] = 0u;                                            // stride msbs / dim1_stride
  g1[7] = 0u;
  asm volatile("tensor_load_to_lds %0, %1" :: "s"(g0), "s"(g1) : "memory");
}

// ---------------------------------------------------------------------------
// Async global -> LDS 16-byte copy (ASYNCcnt-tracked).
// ---------------------------------------------------------------------------
__device__ __forceinline__ void async_copy_b128(u32 lds_off, const float* gsrc) {
  asm volatile("global_load_async_to_lds_b128 %0, %1, off"
               :: "v"(lds_off), "v"(gsrc) : "memory");
}

// ---------------------------------------------------------------------------
// Embedding gather: h[l,e] = emb[tokens[l], e]
// ---------------------------------------------------------------------------
__global__ void embed_kernel(const int* __restrict__ tok, const float* __restrict__ emb,
                             float* __restrict__ h) {
  int idx = blockIdx.x * blockDim.x + threadIdx.x;
  if (idx >= LSEQ * NE) return;
  int l = idx / NE, e = idx % NE;
  h[idx] = emb[(size_t)tok[l] * NE + e];
}

// ---------------------------------------------------------------------------
// RMSNorm over last axis (E=384), one block (4 waves) per row.
// ---------------------------------------------------------------------------
__global__ void rmsnorm_kernel(const float* __restrict__ x, const float* __restrict__ w,
                               float* __restrict__ out) {
  __shared__ float red[4];
  __shared__ float inv_s;
  int row = blockIdx.x;
  const float* xr = x + (size_t)row * NE;
  float ss = 0.f;
  for (int i = threadIdx.x; i < NE; i += blockDim.x) { float v = xr[i]; ss += v * v; }
#pragma unroll
  for (int o = 16; o > 0; o >>= 1) ss += __shfl_down(ss, o, 32);
  if ((threadIdx.x & 31) == 0) red[threadIdx.x >> 5] = ss;
  __syncthreads();
  if (threadIdx.x == 0) {
    float s = red[0] + red[1] + red[2] + red[3];
    inv_s = rsqrtf(s / (float)NE + 1e-5f);
  }
  __syncthreads();
  float inv = inv_s;
  for (int i = threadIdx.x; i < NE; i += blockDim.x)
    out[(size_t)row * NE + i] = xr[i] * inv * w[i];
}

// ---------------------------------------------------------------------------
// WMMA GEMM: out[M,N] = A[M,K] @ W[N,K]^T (+bias) (+= when accumulate!=0)
// Block computes a 16(M) x 64(N) tile; K is walked in KB=32 chunks that are
// staged in LDS with global_load_async_to_lds_b128 (one b128 per lane),
// drained with s_wait_asynccnt + barrier, then consumed by
// V_WMMA_F32_16X16X4_F32 (one 16x16 tile per wave).
// Fragment layouts per CDNA5 ISA 7.12.2:
//   A 16x4: lanes 0-15 -> (K=k, k+1); lanes 16-31 -> (k+2, k+3)
//   B 4x16: lane%16 = column n; same K striping as A
//   C/D 16x16: VGPR j -> row m0+j (lanes<16) / m0+j+8 (lanes>=16), lane%16=col
// grid = (ceil(N/64), M/16), block = 128 (4 waves).
// ---------------------------------------------------------------------------
__global__ void wmma_gemm(const float* __restrict__ A, const float* __restrict__ W,
                          const float* __restrict__ bias, float* __restrict__ out,
                          int N, int K, int accumulate) {
  __shared__ __align__(16) float sA[16 * KB];   // 2 KB
  __shared__ __align__(16) float sW[64 * KB];   // 8 KB
  const int tid  = threadIdx.x;
  const int lane = tid & 31;
  const int wave = tid >> 5;
  const int m0   = blockIdx.y << 4;
  const int n0b  = blockIdx.x << 6;            // block column base
  const int n0   = n0b + (wave << 4);          // this wave's 16 columns
  const int half = lane >> 4;                  // 0: lanes 0-15, 1: lanes 16-31
  const int l16  = lane & 15;
  const int kh   = 2 * half;
  const bool wactive = (n0 < N);               // wave-uniform

  const u32 sA_off = (u32)(unsigned long long)&sA[0];
  const u32 sW_off = (u32)(unsigned long long)&sW[0];

  v8f c = {0.f, 0.f, 0.f, 0.f, 0.f, 0.f, 0.f, 0.f};

  for (int k0 = 0; k0 < K; k0 += KB) {
    // --- stage A tile: 16 rows x 32 floats = 128 x b128, one per thread ---
    {
      int row = tid >> 3;                      // 0..15
      int c4  = (tid & 7) << 2;                // 0,4,...,28
      const float* g = A + (size_t)(m0 + row) * K + k0 + c4;
      __builtin_prefetch(g + KB, 0, 0);        // prefetch next chunk line
      async_copy_b128(sA_off + (u32)((row * KB + c4) << 2), g);
    }
    // --- stage W tile: 64 rows x 32 floats = 512 x b128, four per thread ---
#pragma unroll
    for (int r = 0; r < 4; ++r) {
      int li  = tid + (r << 7);                // 0..511
      int row = li >> 3;                       // 0..63
      int c4  = (li & 7) << 2;
      int wr  = n0b + row;
      if (wr >= N) wr = N - 1;                 // clamp ragged tail (unused data)
      const float* g = W + (size_t)wr * K + k0 + c4;
      async_copy_b128(sW_off + (u32)((row * KB + c4) << 2), g);
    }
    asm volatile("s_wait_asynccnt 0x0" ::: "memory");
    __syncthreads();

    if (wactive) {
      const float* aR = &sA[l16 * KB];                  // A row m0+l16
      const float* wR = &sW[((wave << 4) + l16) * KB];  // B column n0+l16
#pragma unroll
      for (int kk = 0; kk < KB; kk += 4) {
        v2f a, b;
        a.x = aR[kk + kh];
        a.y = aR[kk + kh + 1];
        b.x = wR[kk + kh];
        b.y = wR[kk + kh + 1];
        // (neg_a, A, neg_b, B, c_mod, C, reuse_a, reuse_b)
        c = __builtin_amdgcn_wmma_f32_16x16x4_f32(false, a, false, b, (short)0,
                                                  c, false, false);
      }
    }
    __syncthreads();
  }

  if (!wactive) return;
  float bc = bias ? bias[n0 + l16] : 0.f;
#pragma unroll
  for (int j = 0; j < 8; ++j) {
    int row = m0 + j + 8 * half;
    size_t idx = (size_t)row * N + n0 + l16;
    float v = c[j] + bc;
    out[idx] = accumulate ? (out[idx] + v) : v;
  }
}

// ---------------------------------------------------------------------------
// Causal depthwise conv1d (pad K-1, keep first L) + SiLU.
// ---------------------------------------------------------------------------
__global__ void conv_silu_kernel(const float* __restrict__ xr, const float* __restrict__ cw,
                                 const float* __restrict__ cb, float* __restrict__ xs) {
  int idx = blockIdx.x * blockDim.x + threadIdx.x;
  if (idx >= LSEQ * DI) return;
  int l = idx / DI, d = idx % DI;
  float acc = cb[d];
#pragma unroll
  for (int k = 0; k < KC; ++k) {
    int li = l - (KC - 1) + k;
    if (li >= 0) acc += xr[(size_t)li * (2 * DI) + d] * cw[d * KC + k];
  }
  xs[idx] = acc / (1.f + expf(-acc));  // silu
}

// ---------------------------------------------------------------------------
// delta[l,d] = softplus( dbc[l, 0:R] . dt_w[d, :] )
// ---------------------------------------------------------------------------
__global__ void delta_kernel(const float* __restrict__ dbc, const float* __restrict__ dtw,
                             float* __restrict__ dlt) {
  int idx = blockIdx.x * blockDim.x + threadIdx.x;
  if (idx >= LSEQ * DI) return;
  int l = idx / DI, d = idx % DI;
  const float* row  = dbc + (size_t)l * (RR + 2 * NS);
  const float* wrow = dtw + (size_t)d * RR;
  float acc = 0.f;
#pragma unroll
  for (int r = 0; r < RR; ++r) acc += row[r] * wrow[r];
  dlt[idx] = (acc > 20.f) ? acc : log1pf(expf(acc));
}

// ---------------------------------------------------------------------------
// Selective scan. One wave32 per d-channel (8 waves/block, 96 blocks = 768 d).
// Each lane keeps 24 states (n = lane + 32*j) and the A row in registers.
// The (delta_low|B|C) row for timestep t is DMA'd into LDS by the Tensor
// Data Mover (wave 0 issues, waits TENSORcnt, block barriers).
// ---------------------------------------------------------------------------
__global__ void scan_kernel(const float* __restrict__ dbc, const float* __restrict__ dlt,
                            const float* __restrict__ xs, const float* __restrict__ A_log,
                            const float* __restrict__ Dp, float* __restrict__ y) {
  __shared__ __align__(16) float sRow[RR + 2 * NS];   // 1560 floats
  const int lane = threadIdx.x & 31;
  const int wave = threadIdx.x >> 5;
  const int d = blockIdx.x * 8 + wave;
  const u32 sRow_off = (u32)(unsigned long long)&sRow[0];
  const float* sB = &sRow[RR];
  const float* sC = &sRow[RR + NS];

  float h[NS / 32], Aa[NS / 32];
#pragma unroll
  for (int j = 0; j < NS / 32; ++j) {
    h[j]  = 0.f;
    Aa[j] = -expf(A_log[(size_t)d * NS + lane + 32 * j]);  // A = -exp(A_log)
  }
  const float dskip = Dp[d];

  for (int t = 0; t < LSEQ; ++t) {
    if (wave == 0) {                      // wave-uniform: one TDM per block
      tdm_load_row_f32(dbc + (size_t)t * (RR + 2 * NS), sRow_off, RR + 2 * NS);
      __builtin_amdgcn_s_wait_tensorcnt(0);
    }
    __syncthreads();

    float dl  = dlt[(size_t)t * DI + d];
    float uv  = xs[(size_t)t * DI + d];
    float dlu = dl * uv;
    float acc = 0.f;
#pragma unroll
    for (int j = 0; j < NS / 32; ++j) {
      int n = lane + 32 * j;
      h[j] = expf(dl * Aa[j]) * h[j] + dlu * sB[n];
      acc += h[j] * sC[n];
    }
#pragma unroll
    for (int o = 16; o > 0; o >>= 1) acc += __shfl_down(acc, o, 32);
    if (lane == 0) y[(size_t)t * DI + d] = acc + uv * dskip;
    __syncthreads();
  }
}

// ---------------------------------------------------------------------------
// yg = y * silu(res), res = second half of xr rows.
// ---------------------------------------------------------------------------
__global__ void gate_kernel(const float* __restrict__ y, const float* __restrict__ xr,
                            float* __restrict__ yg) {
  int idx = blockIdx.x * blockDim.x + threadIdx.x;
  if (idx >= LSEQ * DI) return;
  int l = idx / DI, d = idx % DI;
  float r = xr[(size_t)l * (2 * DI) + DI + d];
  yg[idx] = y[idx] * (r / (1.f + expf(-r)));
}

// ---------------------------------------------------------------------------
extern "C" void kernel_launch(void* const* d_in, const int* in_sizes, int n_in,
                              void* d_out, int out_size, void* d_ws, size_t ws_size,
                              hipStream_t stream) {
  (void)in_sizes; (void)n_in; (void)out_size; (void)ws_size;
  const int*   tokens  = (const int*)  d_in[0];
  const float* emb     = (const float*)d_in[1];
  const float* in_w    = (const float*)d_in[2];
  const float* in_b    = (const float*)d_in[3];
  const float* conv_w  = (const float*)d_in[4];
  const float* conv_b  = (const float*)d_in[5];
  const float* ssm_in  = (const float*)d_in[6];
  const float* dt_w    = (const float*)d_in[7];
  const float* A_log   = (const float*)d_in[8];
  const float* Dp      = (const float*)d_in[9];
  const float* out_w   = (const float*)d_in[10];
  const float* out_b   = (const float*)d_in[11];
  const float* norm_w  = (const float*)d_in[12];
  const float* fnorm_w = (const float*)d_in[13];
  float* out = (float*)d_out;

  float* ws  = (float*)d_ws;
  float* h   = ws;  ws += LSEQ * NE;
  float* xn  = ws;  ws += LSEQ * NE;
  float* xr  = ws;  ws += LSEQ * 2 * DI;
  float* xs  = ws;  ws += LSEQ * DI;
  float* dbc = ws;  ws += LSEQ * (RR + 2 * NS);
  float* dlt = ws;  ws += LSEQ * DI;
  float* yb  = ws;  ws += LSEQ * DI;
  float* yg  = ws;  ws += LSEQ * DI;

  embed_kernel<<<(LSEQ * NE + 255) / 256, 256, 0, stream>>>(tokens, emb, h);

  for (int i = 0; i < NLAY; ++i) {
    const float* in_w_i   = in_w   + (size_t)i * 2 * DI * NE;
    const float* in_b_i   = in_b   + (size_t)i * 2 * DI;
    const float* conv_w_i = conv_w + (size_t)i * DI * KC;
    const float* conv_b_i = conv_b + (size_t)i * DI;
    const float* ssm_in_i = ssm_in + (size_t)i * (RR + 2 * NS) * DI;
    const float* dt_w_i   = dt_w   + (size_t)i * DI * RR;
    const float* A_log_i  = A_log  + (size_t)i * DI * NS;
    const float* Dp_i     = Dp     + (size_t)i * DI;
    const float* out_w_i  = out_w  + (size_t)i * NE * DI;
    const float* out_b_i  = out_b  + (size_t)i * NE;
    const float* nw_i     = norm_w + (size_t)i * NE;

    rmsnorm_kernel<<<LSEQ, 128, 0, stream>>>(h, nw_i, xn);

    // xr[L, 2D] = xn @ in_w^T + in_b   (N=1536, K=384)
    wmma_gemm<<<dim3((2 * DI) / 64, LSEQ / 16), 128, 0, stream>>>(
        xn, in_w_i, in_b_i, xr, 2 * DI, NE, 0);

    conv_silu_kernel<<<(LSEQ * DI + 255) / 256, 256, 0, stream>>>(
        xr, conv_w_i, conv_b_i, xs);

    // dbc[L, R+2N] = xs @ ssm_in^T    (N=1560, K=768)
    wmma_gemm<<<dim3((RR + 2 * NS + 63) / 64, LSEQ / 16), 128, 0, stream>>>(
        xs, ssm_in_i, nullptr, dbc, RR + 2 * NS, DI, 0);

    delta_kernel<<<(LSEQ * DI + 255) / 256, 256, 0, stream>>>(dbc, dt_w_i, dlt);

    scan_kernel<<<DI / 8, 256, 0, stream>>>(dbc, dlt, xs, A_log_i, Dp_i, yb);

    gate_kernel<<<(LSEQ * DI + 255) / 256, 256, 0, stream>>>(yb, xr, yg);

    // h += yg @ out_w^T + out_b       (N=384, K=768, residual accumulate)
    wmma_gemm<<<dim3(NE / 64, LSEQ / 16), 128, 0, stream>>>(
        yg, out_w_i, out_b_i, h, NE, DI, 1);
  }

  rmsnorm_kernel<<<LSEQ, 128, 0, stream>>>(h, fnorm_w, xn);

  // logits[L, V] = xn @ emb^T         (N=32000, K=384)
  wmma_gemm<<<dim3(VOC / 64, LSEQ / 16), 128, 0, stream>>>(
      xn, emb, nullptr, out, VOC, NE, 0);
}